// Spatial_network_32710470926816
// MI455X (gfx1250) — compile-verified
//
#include <hip/hip_runtime.h>
#include <hip/hip_bf16.h>

typedef float v2f __attribute__((ext_vector_type(2)));
typedef float v4f __attribute__((ext_vector_type(4)));
typedef float v8f __attribute__((ext_vector_type(8)));

#define B_    8
#define D_IN  256
#define D_HID 1024
#define D_OUT 40000          // = 200*200
#define CHW   (256 * 40000)  // per-batch x elements

// ---------------------------------------------------------------------------
// Kernel 1: hT[j][m] = LeakyReLU(loc[m] . W1[:,j] + b1[j]), m padded 8..15 = 0
// 1024 threads total (4 blocks x 256). W1 row reads are lane-coalesced.
// ---------------------------------------------------------------------------
__global__ void mlp1_kernel(const float* __restrict__ loc,   // [8][256]
                            const float* __restrict__ W1,    // [256][1024]
                            const float* __restrict__ b1,    // [1024]
                            float* __restrict__ hT)          // [1024][16]
{
    __shared__ float sloc[B_ * D_IN];   // 8 KB
    for (int t = threadIdx.x; t < B_ * D_IN; t += blockDim.x)
        sloc[t] = loc[t];
    __syncthreads();

    const int j = blockIdx.x * blockDim.x + threadIdx.x;   // 0..1023
    float acc[B_];
    const float bias = b1[j];
#pragma unroll
    for (int b = 0; b < B_; ++b) acc[b] = bias;

    for (int i = 0; i < D_IN; ++i) {
        const float wv = W1[i * D_HID + j];    // coalesced across lanes
#pragma unroll
        for (int b = 0; b < B_; ++b) acc[b] += sloc[b * D_IN + i] * wv;
    }
#pragma unroll
    for (int b = 0; b < B_; ++b) {
        const float v = acc[b];
        hT[j * 16 + b] = (v >= 0.0f) ? v : 0.1f * v;       // LeakyReLU(0.1)
    }
#pragma unroll
    for (int b = B_; b < 16; ++b) hT[j * 16 + b] = 0.0f;   // pad rows M=8..15
}

// ---------------------------------------------------------------------------
// Kernel 2: w[b][n] = hT . W2 + b2 via V_WMMA_F32_16X16X4_F32.
// One wave per 16-wide N tile (2500 tiles); M=16 covers all 8 batches + pad.
// Each W2 element is read exactly once device-wide -> bound by 164 MB read.
// ---------------------------------------------------------------------------
__global__ void mlp2_wmma_kernel(const float* __restrict__ hT,  // [1024][16]
                                 const float* __restrict__ W2,  // [1024][40000]
                                 const float* __restrict__ b2,  // [40000]
                                 float* __restrict__ w)         // [8][40000]
{
    const int lane  = threadIdx.x & 31;
    const int wave  = threadIdx.x >> 5;
    const int tile  = blockIdx.x * (blockDim.x >> 5) + wave;    // 0..2499 exact
    const int n0    = tile * 16;
    const int m     = lane & 15;
    const int kOff  = (lane >> 4) << 1;   // lanes 0-15 -> K+{0,1}; 16-31 -> K+{2,3}

    v8f acc = {};
#pragma unroll 4
    for (int k = 0; k < D_HID; k += 4) {
        const int ka = k + kOff;
        v2f a, b;
        a.x = hT[(size_t)(ka    ) * 16 + m];
        a.y = hT[(size_t)(ka + 1) * 16 + m];
        b.x = W2[(size_t)(ka    ) * D_OUT + n0 + m];
        b.y = W2[(size_t)(ka + 1) * D_OUT + n0 + m];
        // D = A(16x4,f32) * B(4x16,f32) + C ; exact f32 matrix FMA
        acc = __builtin_amdgcn_wmma_f32_16x16x4_f32(
                  false, a, false, b, (short)0, acc, false, false);
    }

    // D layout: VGPR r, lanes 0-15 -> (M=r, N=n0+lane). M=0..7 are batches.
    if (lane < 16) {
        const float bias = b2[n0 + lane];
#pragma unroll
        for (int r = 0; r < B_; ++r)
            w[(size_t)r * D_OUT + n0 + lane] = acc[r] + bias;
    }
}

// ---------------------------------------------------------------------------
// Kernel 3: out = x * (1 + w[b, p])  — pure streaming, v4f (clang ext-vector),
// NT hints on the 656 MB x/out stream so the 1.28 MB w stays cache-resident.
// ---------------------------------------------------------------------------
__global__ void scale_kernel(const v4f* __restrict__ x,
                             const float* __restrict__ w,
                             v4f* __restrict__ out)
{
    const unsigned idx = blockIdx.x * blockDim.x + threadIdx.x;  // v4f index
    const unsigned e0  = idx * 4u;                               // element index
    const unsigned b   = e0 / (unsigned)CHW;
    const unsigned p   = e0 % (unsigned)D_OUT;

    const v4f wv = *reinterpret_cast<const v4f*>(w + (size_t)b * D_OUT + p);
    const v4f xv = __builtin_nontemporal_load(x + idx);
    v4f o;
    o.x = xv.x * (1.0f + wv.x);
    o.y = xv.y * (1.0f + wv.y);
    o.z = xv.z * (1.0f + wv.z);
    o.w = xv.w * (1.0f + wv.w);
    __builtin_nontemporal_store(o, out + idx);
}

// ---------------------------------------------------------------------------
extern "C" void kernel_launch(void* const* d_in, const int* in_sizes, int n_in,
                              void* d_out, int out_size, void* d_ws, size_t ws_size,
                              hipStream_t stream)
{
    const float* x   = (const float*)d_in[0];   // [8,256,200,200]
    const float* loc = (const float*)d_in[1];   // [8,256]
    const float* W1  = (const float*)d_in[2];   // [256,1024]
    const float* b1  = (const float*)d_in[3];   // [1024]
    const float* W2  = (const float*)d_in[4];   // [1024,40000]
    const float* b2  = (const float*)d_in[5];   // [40000]
    float* out       = (float*)d_out;

    float* hT = (float*)d_ws;                        // 1024*16 floats = 64 KB
    float* w  = hT + (size_t)D_HID * 16;             // 8*40000 floats = 1.28 MB

    // Stage 1: tiny MLP layer (4 blocks x 256 threads)
    mlp1_kernel<<<D_HID / 256, 256, 0, stream>>>(loc, W1, b1, hT);

    // Stage 2: WMMA GEMM. 2500 tiles, 4 waves/block of 128 -> 625 blocks exact.
    mlp2_wmma_kernel<<<(D_OUT / 16) / 4, 128, 0, stream>>>(hT, W2, b2, w);

    // Stage 3: streaming scale. 81,920,000 elems / 4 / 256 = 80,000 blocks exact.
    scale_kernel<<<(B_ * CHW) / 4 / 256, 256, 0, stream>>>(
        (const v4f*)x, w, (v4f*)out);
}